// prj_module_63840393888309
// MI455X (gfx1250) — compile-verified
//
#include <hip/hip_runtime.h>
#include <hip/hip_bf16.h>

#define HH 256
#define WW 256
#define VV 192
#define DD 256
#define SS 256

static __device__ constexpr float SQ2   = 1.41421356237309504880f;
static __device__ constexpr float PI_F  = 3.14159265358979323846f;
static __device__ constexpr float DT    = 2.0f * 1.41421356237309504880f / 256.0f;  // 2*sqrt2/S
static __device__ constexpr float DSTEP = 2.0f * 1.41421356237309504880f / 255.0f;  // det step (D-1)
static __device__ constexpr float TSTEP = 2.0f * 1.41421356237309504880f / 255.0f;  // t step (S-1)

// Sentinel index: far outside the 256KB LDS allocation -> HW returns 0 / drops store.
// (All iy out-of-range cases land OOB by construction; only x needs an explicit select.)
#define OOB_IDX 0x01000000u

typedef __attribute__((ext_vector_type(2))) float v2f;
typedef __attribute__((ext_vector_type(8))) float v8f;

// ---------- helpers ----------

struct BL { unsigned i00, i10, i01, i11; float fx, fy; };

__device__ __forceinline__ BL bl_prep(float xp, float yp) {
    BL r;
    float x0 = floorf(xp), y0 = floorf(yp);
    int ix = (int)x0, iy = (int)y0;
    r.fx = xp - x0;  r.fy = yp - y0;
    unsigned row0 = (unsigned)(iy * WW);          // iy<0 or iy>=256 -> OOB byte addr
    unsigned row1 = row0 + (unsigned)WW;
    unsigned cx0 = ((unsigned)ix       < (unsigned)WW) ? (unsigned)ix       : OOB_IDX;
    unsigned cx1 = ((unsigned)(ix + 1) < (unsigned)WW) ? (unsigned)(ix + 1) : OOB_IDX;
    r.i00 = row0 + cx0;  r.i10 = row0 + cx1;
    r.i01 = row1 + cx0;  r.i11 = row1 + cx1;
    return r;
}

__device__ __forceinline__ float bl_mix(float v00, float v10, float v01, float v11,
                                        float fx, float fy) {
    float a = fmaf(fx, v10 - v00, v00);
    float b = fmaf(fx, v11 - v01, v01);
    return fmaf(fy, b - a, a);
}

// CDNA5 async global->LDS copy of 16 bytes (tracked with ASYNCcnt)
__device__ __forceinline__ void async_g2l_b128(unsigned lds_byte, int gbl_voff,
                                               unsigned long long gbl_base) {
    asm volatile("global_load_async_to_lds_b128 %0, %1, %2 offset:0"
                 :: "v"(lds_byte), "v"(gbl_voff), "s"(gbl_base)
                 : "memory");
}

__device__ __forceinline__ void wait_async0() {
    asm volatile("s_wait_asynccnt 0" ::: "memory");
}

// ---------- kernels ----------

__global__ void zero_kernel(float* p, int n) {
    int i = blockIdx.x * blockDim.x + threadIdx.x;
    if (i < n) p[i] = 0.0f;
}

// Forward projection: grid (V, B), block 256.  stemp[b,v,d] = sino - proj
__global__ __launch_bounds__(256)
void fwd_proj_kernel(const float* __restrict__ img_all,
                     const float* __restrict__ proj_all,
                     float* __restrict__ stemp) {
    extern __shared__ float smem[];            // HH*WW floats = 256 KB (LDS is 320 KB/WGP)
    const int v   = blockIdx.x;
    const int b   = blockIdx.y;
    const int tid = threadIdx.x;

    // --- stage whole image into LDS via async global->LDS b128 ---
    const unsigned lds_base = (unsigned)(size_t)smem;
    const unsigned long long gbase =
        (unsigned long long)(const void*)(img_all + (size_t)b * HH * WW);
    #pragma unroll 4
    for (int it = 0; it < (HH * WW * 4) / (256 * 16); ++it) {   // 64 iters * 4KB
        int boff = (tid + it * 256) * 16;
        async_g2l_b128(lds_base + (unsigned)boff, boff, gbase);
    }
    wait_async0();
    __syncthreads();

    // --- per-ray accumulation; S-reduction on the WMMA pipe ---
    const int lane = tid & 31;
    const int wave = tid >> 5;                 // 8 waves of 32 (wave32)
    const int m    = lane & 15;                // detector row in 16x16 tile
    const int kb   = (lane >> 4) * 2;          // K: lanes 0-15 -> k=0,1 ; 16-31 -> k=2,3

    const float ang = (PI_F / (float)VV) * (float)v;
    const float ca  = cosf(ang);
    const float sa  = sinf(ang);

    // fold t->world->pixel chain into  xp = Ax + si*Kx  (one add per sample)
    const float Kx  = TSTEP * ca * (WW * 0.5f);
    const float Ky  = TSTEP * sa * (HH * 0.5f);
    const float Kx4 = 4.0f * Kx, Ky4 = 4.0f * Ky;

    v2f bones; bones.x = 1.0f; bones.y = 1.0f; // B = ones(4x16) -> row sums

    for (int half = 0; half < 2; ++half) {
        const int   dbase = half * 128 + wave * 16;
        const int   d     = dbase + m;
        const float det   = -SQ2 + (float)d * DSTEP;
        const float bx    = -det * sa;         // xw = bx + t*ca
        const float by    =  det * ca;         // yw = by + t*sa
        const float Ax = (bx - SQ2 * ca + 1.0f) * (WW * 0.5f) - 0.5f;  // si = 0
        const float Ay = (by - SQ2 * sa + 1.0f) * (HH * 0.5f) - 0.5f;

        float xp0 = fmaf((float)kb, Kx, Ax);
        float yp0 = fmaf((float)kb, Ky, Ay);

        v8f c = {};                            // 16x16 f32 accumulator tile
        #pragma unroll 4
        for (int s0 = 0; s0 < SS; s0 += 4) {
            // all 8 corner addresses first, then the 8 ds_loads, then math
            BL p0 = bl_prep(xp0,      yp0);
            BL p1 = bl_prep(xp0 + Kx, yp0 + Ky);
            float q00 = smem[p0.i00], q10 = smem[p0.i10];
            float q01 = smem[p0.i01], q11 = smem[p0.i11];
            float r00 = smem[p1.i00], r10 = smem[p1.i10];
            float r01 = smem[p1.i01], r11 = smem[p1.i11];
            v2f a;
            a.x = bl_mix(q00, q10, q01, q11, p0.fx, p0.fy);
            a.y = bl_mix(r00, r10, r01, r11, p1.fx, p1.fy);
            // D = A(16x4) x ones(4x16) + C  ->  every column of C holds row sums
            c = __builtin_amdgcn_wmma_f32_16x16x4_f32(false, a, false, bones,
                                                      (short)0, c, false, false);
            xp0 += Kx4;  yp0 += Ky4;
        }

        // column N=0 lives in lane 0 (rows 0..7) and lane 16 (rows 8..15)
        if (m == 0) {
            const int rowbase = (lane >> 4) * 8;
            #pragma unroll
            for (int r = 0; r < 8; ++r) {
                const int dd  = dbase + rowbase + r;
                const int idx = (b * VV + v) * DD + dd;
                stemp[idx] = c[r] * DT - proj_all[idx];
            }
        }
    }
}

// Adjoint (exact bilinear scatter): grid (V, B), block 256 (thread == detector).
// Branchless: invalid corners are steered to an OOB LDS address; HW drops them,
// which is exactly the adjoint of the gather's zero-outside behavior.
__global__ __launch_bounds__(256)
void back_proj_kernel(const float* __restrict__ stemp,
                      float* __restrict__ ir) {
    extern __shared__ float acc[];             // HH*WW floats accumulator
    const int v   = blockIdx.x;
    const int b   = blockIdx.y;
    const int tid = threadIdx.x;

    for (int i = tid; i < HH * WW; i += 256) acc[i] = 0.0f;
    __syncthreads();

    const float ang = (PI_F / (float)VV) * (float)v;
    const float ca  = cosf(ang);
    const float sa  = sinf(ang);

    const int   d    = tid;
    const float tval = stemp[(b * VV + v) * DD + d] * DT;
    const float det  = -SQ2 + (float)d * DSTEP;
    const float bx   = -det * sa;
    const float by   =  det * ca;

    const float Kx = TSTEP * ca * (WW * 0.5f);
    const float Ky = TSTEP * sa * (HH * 0.5f);
    float xp = (bx - SQ2 * ca + 1.0f) * (WW * 0.5f) - 0.5f;
    float yp = (by - SQ2 * sa + 1.0f) * (HH * 0.5f) - 0.5f;

    for (int si = 0; si < SS; ++si) {
        BL p = bl_prep(xp, yp);
        float gx  = 1.0f - p.fx, gy = 1.0f - p.fy;
        float wy0 = tval * gy,   wy1 = tval * p.fy;
        atomicAdd(&acc[p.i00], gx   * wy0);    // ds_add_f32 (OOB dropped by HW)
        atomicAdd(&acc[p.i10], p.fx * wy0);
        atomicAdd(&acc[p.i01], gx   * wy1);
        atomicAdd(&acc[p.i11], p.fx * wy1);
        xp += Kx;  yp += Ky;
    }
    __syncthreads();

    float* irb = ir + (size_t)b * HH * WW;
    for (int i = tid; i < HH * WW; i += 256)
        atomicAdd(&irb[i], acc[i]);            // global_atomic_add_f32
}

// out = in - w * ir
__global__ void axpy_kernel(const float* __restrict__ in,
                            const float* __restrict__ ir,
                            const float* __restrict__ w,
                            float* __restrict__ out, int n) {
    int i = blockIdx.x * blockDim.x + threadIdx.x;
    if (i < n) out[i] = in[i] - w[0] * ir[i];
}

// ---------- launcher ----------

extern "C" void kernel_launch(void* const* d_in, const int* in_sizes, int n_in,
                              void* d_out, int out_size, void* d_ws, size_t ws_size,
                              hipStream_t stream) {
    const float* input  = (const float*)d_in[0];   // (2,1,256,256) f32
    const float* proj   = (const float*)d_in[1];   // (2,1,192,256) f32
    const float* weight = (const float*)d_in[2];   // (1,) f32

    float* ws    = (float*)d_ws;
    float* ir    = ws;                  // 2*H*W floats
    float* stemp = ws + 2 * HH * WW;    // 2*V*D floats

    const int nimg = 2 * HH * WW;
    zero_kernel<<<(nimg + 255) / 256, 256, 0, stream>>>(ir, nimg);

    dim3 grid(VV, 2);
    const size_t lds_bytes = (size_t)HH * WW * sizeof(float);   // 256 KB of 320 KB WGP LDS
    fwd_proj_kernel <<<grid, 256, lds_bytes, stream>>>(input, proj, stemp);
    back_proj_kernel<<<grid, 256, lds_bytes, stream>>>(stemp, ir);

    axpy_kernel<<<(nimg + 255) / 256, 256, 0, stream>>>(input, ir, weight,
                                                        (float*)d_out, nimg);
}